// GlmLayer_39874476376639
// MI455X (gfx1250) — compile-verified
//
#include <hip/hip_runtime.h>
#include <hip/hip_bf16.h>
#include <math.h>

// ---------------- problem constants ----------------
#define BB   128
#define SS   512
#define HD   2048
#define NH   16
#define DN   128
#define DR   64
#define DV   128
#define II   1408
#define EE   32
#define KTOP 8
#define EPS  1e-6f

#define QSTRIDE  (NH * (DN + DR))          // 3072
#define KVSTRIDE (NH * (DN + DV) + DR)     // 4160

typedef float v2f __attribute__((ext_vector_type(2)));
typedef float v8f __attribute__((ext_vector_type(8)));

// =====================================================================
// GEMM: C[M,N] = A[M,K] * W[N,K]^T   (A row-major lda, W row-major ldw)
// One wave computes a 16x16 tile using V_WMMA_F32_16X16X4_F32.
// A-frag (16x4): lanes 0-15 -> M=lane, {K=k+0,k+1}; lanes 16-31 -> {k+2,k+3}
// B-frag (4x16): lanes 0-15 -> N=lane, {K=k+0,k+1}; lanes 16-31 -> {k+2,k+3}
// D (16x16): acc[r] = row (r + 8*(lane>>4)), col (lane&15)
// =====================================================================
template <bool ATOMIC>
__global__ __launch_bounds__(256) void gemm_xt_wmma(
    const float* __restrict__ A, int lda,
    const float* __restrict__ W, int ldw,
    float* __restrict__ C, int ldc,
    int M, int N, int Kdim)
{
    const int wave = threadIdx.x >> 5;
    const int lane = threadIdx.x & 31;
    const int nt   = blockIdx.x * 8 + wave;   // N tile index
    if (nt * 16 >= N) return;                 // wave-uniform exit (EXEC stays full)
    const int mt   = blockIdx.y;              // M tile index
    const int half = lane >> 4;
    const int l16  = lane & 15;

    const float* __restrict__ arow = A + (size_t)(mt * 16 + l16) * lda + 2 * half;
    const float* __restrict__ wrow = W + (size_t)(nt * 16 + l16) * ldw + 2 * half;

    v8f acc = {0.f, 0.f, 0.f, 0.f, 0.f, 0.f, 0.f, 0.f};

    for (int k = 0; k < Kdim; k += 4) {
        v2f a = *(const v2f*)(arow + k);   // 8-byte global_load_b64
        v2f b = *(const v2f*)(wrow + k);
        acc = __builtin_amdgcn_wmma_f32_16x16x4_f32(
            /*neg_a=*/false, a, /*neg_b=*/false, b,
            /*c_mod=*/(short)0, acc, /*reuse_a=*/false, /*reuse_b=*/false);
    }

    float* crow = C + (size_t)(mt * 16 + 8 * half) * ldc + nt * 16 + l16;
#pragma unroll
    for (int r = 0; r < 8; ++r) {
        if (ATOMIC) atomicAdd(crow + (size_t)r * ldc, acc[r]);
        else        crow[(size_t)r * ldc] = acc[r];
    }
}

// =====================================================================
// RMSNorm: one block per row of HD floats
// =====================================================================
__global__ __launch_bounds__(256) void rmsnorm_kernel(
    const float* __restrict__ x, const float* __restrict__ w,
    float* __restrict__ out)
{
    const int b   = blockIdx.x;
    const int tid = threadIdx.x;
    const float* xr = x + (size_t)b * HD;

    float ss = 0.f;
    for (int i = tid; i < HD; i += 256) { float t = xr[i]; ss += t * t; }
    for (int off = 16; off; off >>= 1) ss += __shfl_xor(ss, off, 32);

    __shared__ float red[8];
    const int w8 = tid >> 5, lane = tid & 31;
    if (lane == 0) red[w8] = ss;
    __syncthreads();
    __shared__ float rr;
    if (tid == 0) {
        float s = 0.f;
        for (int i = 0; i < 8; ++i) s += red[i];
        rr = rsqrtf(s / (float)HD + EPS);
    }
    __syncthreads();
    const float r = rr;
    for (int i = tid; i < HD; i += 256)
        out[(size_t)b * HD + i] = xr[i] * r * w[i];
}

// =====================================================================
// Attention: one block per (b, h). Scores -> softmax -> p*V.
// Fresh k/v from the kv projection replace cache position S-1.
// =====================================================================
__global__ __launch_bounds__(256) void attn_kernel(
    const float* __restrict__ q,   // [B, QSTRIDE]
    const float* __restrict__ kv,  // [B, KVSTRIDE]
    const float* __restrict__ kc,  // [B,S,NH,DN]
    const float* __restrict__ vc,  // [B,S,NH,DV]
    float* __restrict__ attn)      // [B, NH*DV]
{
    const int bh = blockIdx.x;
    const int b  = bh / NH;
    const int h  = bh % NH;
    const int tid  = threadIdx.x;
    const int wv   = tid >> 5;
    const int lane = tid & 31;

    __shared__ float sc[SS];
    __shared__ float red[16];
    __shared__ float part[256];

    const float* qp   = q  + (size_t)b * QSTRIDE + h * (DN + DR);
    const float* knew = kv + (size_t)b * KVSTRIDE + h * DN;
    const float* vnew = kv + (size_t)b * KVSTRIDE + NH * DN + h * DV;
    const float scale = rsqrtf((float)DN);

    // q fragment: lane holds q[4*lane .. 4*lane+3]  (DN = 32*4)
    float4 qf = *(const float4*)(qp + 4 * lane);

    // ---- scores ----
    for (int s = wv; s < SS; s += 8) {
        const float* krow = (s == SS - 1)
            ? knew : (kc + (((size_t)b * SS + s) * NH + h) * DN);
        float4 kf = *(const float4*)(krow + 4 * lane);
        float p = qf.x * kf.x + qf.y * kf.y + qf.z * kf.z + qf.w * kf.w;
        for (int off = 16; off; off >>= 1) p += __shfl_xor(p, off, 32);
        if (lane == 0) sc[s] = p * scale;
    }
    __syncthreads();

    // ---- softmax: max ----
    float mx = -1e30f;
    for (int s = tid; s < SS; s += 256) mx = fmaxf(mx, sc[s]);
    for (int off = 16; off; off >>= 1) mx = fmaxf(mx, __shfl_xor(mx, off, 32));
    if (lane == 0) red[wv] = mx;
    __syncthreads();
    if (tid == 0) {
        float m2 = red[0];
        for (int i = 1; i < 8; ++i) m2 = fmaxf(m2, red[i]);
        red[8] = m2;
    }
    __syncthreads();
    mx = red[8];

    // ---- softmax: exp + sum ----
    float sum = 0.f;
    for (int s = tid; s < SS; s += 256) { float e = __expf(sc[s] - mx); sc[s] = e; sum += e; }
    __syncthreads();
    for (int off = 16; off; off >>= 1) sum += __shfl_xor(sum, off, 32);
    if (lane == 0) red[wv] = sum;
    __syncthreads();
    if (tid == 0) {
        float s2 = 0.f;
        for (int i = 0; i < 8; ++i) s2 += red[i];
        red[9] = s2;
    }
    __syncthreads();
    const float inv = 1.0f / red[9];

    // ---- p * V : thread owns dim d, half of s-range ----
    const int d  = tid & 127;
    const int sh = tid >> 7;
    float acc = 0.f;
    for (int s = sh * 256; s < sh * 256 + 256; ++s) {
        const float* vrow = (s == SS - 1)
            ? vnew : (vc + (((size_t)b * SS + s) * NH + h) * DV);
        acc += sc[s] * vrow[d];
    }
    part[tid] = acc;
    __syncthreads();
    if (tid < 128)
        attn[(size_t)b * (NH * DV) + h * DV + d] = (part[tid] + part[tid + 128]) * inv;
}

// =====================================================================
// Top-8 softmax router -> dense w_full[B,E]. One wave per row.
// =====================================================================
__global__ __launch_bounds__(32) void topk_kernel(
    const float* __restrict__ logits, float* __restrict__ wfull)
{
    const int b    = blockIdx.x;
    const int lane = threadIdx.x;
    float cur = logits[b * EE + lane];

    float vals[KTOP]; int ids[KTOP];
#pragma unroll
    for (int i = 0; i < KTOP; ++i) {
        float m = cur; int mi = lane;
        for (int off = 16; off; off >>= 1) {
            float om = __shfl_xor(m, off, 32);
            int   oi = __shfl_xor(mi, off, 32);
            if (om > m || (om == m && oi < mi)) { m = om; mi = oi; }
        }
        vals[i] = m; ids[i] = mi;             // identical on all lanes
        if (lane == mi) cur = -1e30f;
    }
    const float mx = vals[0];
    float s = 0.f, ex[KTOP];
#pragma unroll
    for (int i = 0; i < KTOP; ++i) { ex[i] = __expf(vals[i] - mx); s += ex[i]; }
    float outv = 0.f;
#pragma unroll
    for (int i = 0; i < KTOP; ++i) if (ids[i] == lane) outv = ex[i] / s;
    wfull[b * EE + lane] = outv;
}

// =====================================================================
// Elementwise helpers
// =====================================================================
__global__ void fill_zero(float* __restrict__ p, int n)
{
    int i = blockIdx.x * blockDim.x + threadIdx.x;
    if (i < n) p[i] = 0.f;
}

__global__ void add2(const float* __restrict__ a, const float* __restrict__ b,
                     float* __restrict__ o, int n)
{
    int i = blockIdx.x * blockDim.x + threadIdx.x;
    if (i < n) o[i] = a[i] + b[i];
}

// m = rowScale(b) * silu(g) * u   over [B, II] flat; rowScale = wfull[b*E + e] or 1
__global__ void silu_mul(const float* __restrict__ g, const float* __restrict__ u,
                         const float* __restrict__ wfull, int e,
                         float* __restrict__ m, int n)
{
    int i = blockIdx.x * blockDim.x + threadIdx.x;
    if (i >= n) return;
    float gv = g[i];
    float sg = gv / (1.f + __expf(-gv));
    float sc = wfull ? wfull[(i / II) * EE + e] : 1.f;
    m[i] = sc * sg * u[i];
}

// =====================================================================
// Host orchestration
// =====================================================================
static inline void launch_gemm(const float* A, int lda, const float* W, int ldw,
                               float* C, int ldc, int M, int N, int Kdim,
                               bool atomic, hipStream_t stream)
{
    dim3 grid((N / 16 + 7) / 8, M / 16);
    dim3 block(256);
    if (atomic)
        hipLaunchKernelGGL((gemm_xt_wmma<true>),  grid, block, 0, stream,
                           A, lda, W, ldw, C, ldc, M, N, Kdim);
    else
        hipLaunchKernelGGL((gemm_xt_wmma<false>), grid, block, 0, stream,
                           A, lda, W, ldw, C, ldc, M, N, Kdim);
}

extern "C" void kernel_launch(void* const* d_in, const int* in_sizes, int n_in,
                              void* d_out, int out_size, void* d_ws, size_t ws_size,
                              hipStream_t stream)
{
    (void)in_sizes; (void)n_in; (void)out_size; (void)ws_size;

    const float* x         = (const float*)d_in[0];   // [B,1,HD]
    const float* ln1_w     = (const float*)d_in[1];
    const float* ln2_w     = (const float*)d_in[2];
    const float* wq        = (const float*)d_in[3];   // [QSTRIDE, HD]
    const float* wkv       = (const float*)d_in[4];   // [KVSTRIDE, HD]
    const float* wo        = (const float*)d_in[5];   // [HD, NH*DV]
    const float* k_cache   = (const float*)d_in[6];
    const float* v_cache   = (const float*)d_in[7];
    const float* w_router  = (const float*)d_in[8];   // [E, HD]
    const float* w_gate    = (const float*)d_in[9];   // [E, I, HD]
    const float* w_up      = (const float*)d_in[10];  // [E, I, HD]
    const float* w_down    = (const float*)d_in[11];  // [E, HD, I]
    const float* w_sh_gate = (const float*)d_in[12];  // [I, HD]
    const float* w_sh_up   = (const float*)d_in[13];  // [I, HD]
    const float* w_sh_down = (const float*)d_in[14];  // [HD, I]
    /* d_in[15] = positions (unused: always S-1) */

    float* out = (float*)d_out;                       // [B,1,HD]

    // ---- workspace carve-up (floats) ----
    float* f       = (float*)d_ws;
    float* h       = f;                  f += (size_t)BB * HD;        // 262144
    float* qb      = f;                  f += (size_t)BB * QSTRIDE;   // 393216
    float* kvb     = f;                  f += (size_t)BB * KVSTRIDE;  // 532480
    float* attn    = f;                  f += (size_t)BB * HD;
    float* attnout = f;                  f += (size_t)BB * HD;
    float* hidden  = f;                  f += (size_t)BB * HD;
    float* x2d     = f;                  f += (size_t)BB * HD;
    float* logits  = f;                  f += (size_t)BB * EE;
    float* wfull   = f;                  f += (size_t)BB * EE;
    float* gb      = f;                  f += (size_t)BB * II;
    float* ub      = f;                  f += (size_t)BB * II;
    float* mb      = f;                  f += (size_t)BB * II;
    float* accb    = f;                  f += (size_t)BB * HD;

    const int nBH  = BB * HD;   // 262144
    const int nBI  = BB * II;   // 180224
    dim3 e256(256);

    // 1) h = rmsnorm(x, ln1_w)
    hipLaunchKernelGGL(rmsnorm_kernel, dim3(BB), e256, 0, stream, x, ln1_w, h);

    // 2) q = h * wq^T ; kv = h * wkv^T
    launch_gemm(h, HD, wq,  HD, qb,  QSTRIDE,  BB, QSTRIDE,  HD, false, stream);
    launch_gemm(h, HD, wkv, HD, kvb, KVSTRIDE, BB, KVSTRIDE, HD, false, stream);

    // 3) attention over cache (+fresh k/v at s = S-1)
    hipLaunchKernelGGL(attn_kernel, dim3(BB * NH), e256, 0, stream,
                       qb, kvb, k_cache, v_cache, attn);

    // 4) attn_out = attn * wo^T ; hidden = x + attn_out
    launch_gemm(attn, NH * DV, wo, NH * DV, attnout, HD, BB, HD, NH * DV, false, stream);
    hipLaunchKernelGGL(add2, dim3((nBH + 255) / 256), e256, 0, stream,
                       x, attnout, hidden, nBH);

    // 5) x2d = rmsnorm(hidden, ln2_w)
    hipLaunchKernelGGL(rmsnorm_kernel, dim3(BB), e256, 0, stream, hidden, ln2_w, x2d);

    // 6) router -> w_full
    launch_gemm(x2d, HD, w_router, HD, logits, EE, BB, EE, HD, false, stream);
    hipLaunchKernelGGL(topk_kernel, dim3(BB), dim3(32), 0, stream, logits, wfull);

    // 7) acc = 0
    hipLaunchKernelGGL(fill_zero, dim3((nBH + 255) / 256), e256, 0, stream, accb, nBH);

    // 8) dense MoE: per expert e, acc += (wfull[:,e] * silu(g)*u) * w_down[e]^T
    for (int e = 0; e < EE; ++e) {
        const float* wg = w_gate + (size_t)e * II * HD;
        const float* wu = w_up   + (size_t)e * II * HD;
        const float* wd = w_down + (size_t)e * HD * II;
        launch_gemm(x2d, HD, wg, HD, gb, II, BB, II, HD, false, stream);
        launch_gemm(x2d, HD, wu, HD, ub, II, BB, II, HD, false, stream);
        hipLaunchKernelGGL(silu_mul, dim3((nBI + 255) / 256), e256, 0, stream,
                           gb, ub, wfull, e, mb, nBI);
        launch_gemm(mb, II, wd, II, accb, HD, BB, HD, II, true, stream);
    }

    // 9) shared expert: acc += silu(x2d*wshg^T) * (x2d*wshu^T) * wshd^T
    launch_gemm(x2d, HD, w_sh_gate, HD, gb, II, BB, II, HD, false, stream);
    launch_gemm(x2d, HD, w_sh_up,   HD, ub, II, BB, II, HD, false, stream);
    hipLaunchKernelGGL(silu_mul, dim3((nBI + 255) / 256), e256, 0, stream,
                       gb, ub, (const float*)nullptr, 0, mb, nBI);
    launch_gemm(mb, II, w_sh_down, II, accb, HD, BB, HD, II, true, stream);

    // 10) out = hidden + acc
    hipLaunchKernelGGL(add2, dim3((nBH + 255) / 256), e256, 0, stream,
                       hidden, accb, out, nBH);
}